// NeuralMemory_87754771791892
// MI455X (gfx1250) — compile-verified
//
#include <hip/hip_runtime.h>

// ---------------------------------------------------------------------------
// CDNA5 (gfx1250) persistent neural-memory trainer, round 2.
//  - f32 WMMA (V_WMMA_F32_16X16X4_F32) for all GEMMs
//  - 10 device-wide barriers per optimizer step (norm partials folded into
//    producer phases; final phase does redundant per-block deterministic
//    reduction of ||g||^2 and the theta/eta/alpha gates, then updates)
//  - gW1/gW2 (dominant wgrad) use LDS-blocked panels staged with
//    global_load_async_to_lds_b128 (ASYNCcnt) and consumed via ds loads
// ---------------------------------------------------------------------------

typedef __attribute__((ext_vector_type(2))) float v2f;
typedef __attribute__((ext_vector_type(8))) float v8f;

#define NBLK 64
#define NTHR 256
#define NTOT (NBLK * NTHR)
#define NWAVE (NTOT / 32)

#define DM  1024   // D_MODEL
#define HD  256    // D_HEAD
#define HID 1024   // hidden (_H)
#define CHK 64     // CHUNK rows
#define NCH 16     // chunks per batch
#define NB  4      // batches
#define EPSF 1e-7f

// ---- flat parameter layout ----
#define OFF_W0 0
#define OFF_B0 (OFF_W0 + HID * HD)
#define OFF_W1 (OFF_B0 + HID)
#define OFF_B1 (OFF_W1 + HID * HID)
#define OFF_W2 (OFF_B1 + HID)
#define OFF_B2 (OFF_W2 + HID * HID)
#define OFF_WO (OFF_B2 + HID)
#define OFF_BO (OFF_WO + HD * HID)
#define OFF_G  (OFF_BO + HD)
#define P_TOTAL (OFF_G + HD)            // 2625024
#define P_MLP  (P_TOTAL - HD)

// ---- grad-norm partial slots (written every step before UPD phase) ----
#define SL_WO 0                         // 1024 tiles
#define SL_W2 (SL_WO + 1024)            // 4096 tiles
#define SL_W1 (SL_W2 + 4096)            // 4096 tiles
#define SL_W0 (SL_W1 + 4096)            // 1024 tiles
#define SL_BO (SL_W0 + 1024)            // 256
#define SL_G  (SL_BO + HD)              // 256
#define SL_B2 (SL_G + HD)               // 1024
#define SL_B1 (SL_B2 + HID)             // 1024
#define SL_B0 (SL_B1 + HID)             // 1024
#define NPART (SL_B0 + HID)             // 13824

// ---- workspace (float offsets, after 256B barrier header) ----
#define ACT (CHK * HID)
#define WS_FLAT 0
#define WS_MOM  (WS_FLAT + P_TOTAL)
#define WS_GRAD (WS_MOM + P_TOTAL)
#define WS_Z0   (WS_GRAD + P_TOTAL)
#define WS_H0   (WS_Z0 + ACT)
#define WS_Z1   (WS_H0 + ACT)
#define WS_H1   (WS_Z1 + ACT)
#define WS_Z2   (WS_H1 + ACT)
#define WS_H2   (WS_Z2 + ACT)
#define WS_Y    (WS_H2 + ACT)
#define WS_DY   (WS_Y  + CHK * HD)
#define WS_GU   (WS_DY + CHK * HD)
#define WS_DZ2  (WS_GU + CHK * HD)
#define WS_DZ1  (WS_DZ2 + ACT)
#define WS_DZ0  (WS_DZ1 + ACT)
#define WS_PART (WS_DZ0 + ACT)          // NPART slots

// LDS panel strides (floats): padded so the two K-half rows of a fragment
// load map to disjoint 16-bank groups (stride*4 % 256B == 64B).
#define AP_LD 144                       // 64 x 144 f32 = 36864 B
#define BP_LD 80                        // 64 x  80 f32 = 20480 B

__device__ __forceinline__ float sigm(float x) { return 1.f / (1.f + __expf(-x)); }
__device__ __forceinline__ float silu(float z) { return z * sigm(z); }
__device__ __forceinline__ float dsilu(float z) {
  float s = sigm(z);
  return s * (1.f + z * (1.f - s));
}

__device__ __forceinline__ v8f wmma4(v2f a, v2f b, v8f c) {
  return __builtin_amdgcn_wmma_f32_16x16x4_f32(false, a, false, b, (short)0, c,
                                               false, false);
}

// Async global->LDS 16B per lane (ASYNCcnt). LDS byte offset is the low 32
// bits of the flat shared-aperture address.
__device__ __forceinline__ void async_g2l_b128(void* ldsp, const void* gsrc) {
  unsigned off = (unsigned)(size_t)ldsp;
  asm volatile("global_load_async_to_lds_b128 %0, %1, off"
               :: "v"(off), "v"(gsrc)
               : "memory");
}
__device__ __forceinline__ void wait_async0() {
  asm volatile("s_wait_asynccnt 0x0" ::: "memory");
}

// Device-wide sense-reversal barrier. bar[0]=count, bar[1]=sense.
__device__ __forceinline__ void grid_barrier(unsigned* bar, unsigned* lbar) {
  __syncthreads();
  if (threadIdx.x == 0) {
    unsigned ls = *lbar ^ 1u;
    *lbar = ls;
    __threadfence();
    if (atomicAdd(&bar[0], 1u) == (unsigned)gridDim.x - 1u) {
      bar[0] = 0u;
      __threadfence();
      *(volatile unsigned*)&bar[1] = ls;
    } else {
      while (*(volatile unsigned*)&bar[1] != ls) __builtin_amdgcn_s_sleep(1);
    }
    __threadfence();
  }
  __syncthreads();
}

// Deterministic block tree-reduction (returns sum to all threads).
__device__ __forceinline__ float block_reduce(float v, float* sred) {
  sred[threadIdx.x] = v;
  __syncthreads();
  for (int s = NTHR / 2; s > 0; s >>= 1) {
    if (threadIdx.x < (unsigned)s) sred[threadIdx.x] += sred[threadIdx.x + s];
    __syncthreads();
  }
  float r = sred[0];
  __syncthreads();
  return r;
}

// Deterministic wave32 butterfly sum (same value in all lanes).
__device__ __forceinline__ float wave_sum(float v) {
#pragma unroll
  for (int m = 16; m >= 1; m >>= 1) v += __shfl_xor(v, m, 32);
  return v;
}

// ---- wave-level 16x16 tile GEMMs (fragment layouts per ISA 7.12.2) ------

// Z/H tile of (A[64xK] @ B[NxK]^T + bias), H = silu(Z).
__device__ __forceinline__ void tile_fwd(const float* __restrict__ A, int lda,
                                         const float* __restrict__ B, int ldb,
                                         int K, int mt, int nt,
                                         const float* __restrict__ bias,
                                         float* __restrict__ Z,
                                         float* __restrict__ H, int ldc) {
  int lane = threadIdx.x & 31;
  int mr = lane & 15;
  int kh = (lane >> 4) << 1;
  const float* pA = A + (size_t)(mt * 16 + mr) * lda;
  const float* pB = B + (size_t)(nt * 16 + mr) * ldb;
  v8f acc = {};
  for (int k = 0; k < K; k += 4) {
    v2f a, b;
    a.x = pA[k + kh]; a.y = pA[k + kh + 1];
    b.x = pB[k + kh]; b.y = pB[k + kh + 1];
    acc = wmma4(a, b, acc);
  }
  int col = nt * 16 + mr;
  int rb = mt * 16 + ((lane >> 4) << 3);
  float bv = bias[col];
#pragma unroll
  for (int i = 0; i < 8; ++i) {
    float z = acc[i] + bv;
    Z[(size_t)(rb + i) * ldc + col] = z;
    H[(size_t)(rb + i) * ldc + col] = silu(z);
  }
}

// Output-layer tile: Y = resid + A @ B^T + bias.
__device__ __forceinline__ void tile_fwd_out(
    const float* __restrict__ A, int lda, const float* __restrict__ B, int ldb,
    int K, int mt, int nt, const float* __restrict__ bias,
    const float* __restrict__ resid, int ldr, float* __restrict__ Y, int ldc) {
  int lane = threadIdx.x & 31;
  int mr = lane & 15;
  int kh = (lane >> 4) << 1;
  const float* pA = A + (size_t)(mt * 16 + mr) * lda;
  const float* pB = B + (size_t)(nt * 16 + mr) * ldb;
  v8f acc = {};
  for (int k = 0; k < K; k += 4) {
    v2f a, b;
    a.x = pA[k + kh]; a.y = pA[k + kh + 1];
    b.x = pB[k + kh]; b.y = pB[k + kh + 1];
    acc = wmma4(a, b, acc);
  }
  int col = nt * 16 + mr;
  int rb = mt * 16 + ((lane >> 4) << 3);
  float bv = bias[col];
#pragma unroll
  for (int i = 0; i < 8; ++i) {
    int row = rb + i;
    Y[(size_t)row * ldc + col] = acc[i] + bv + resid[(size_t)row * ldr + col];
  }
}

// dgrad tile: DZ = (A[64xK] @ B[KxN]) * silu'(Zc).
__device__ __forceinline__ void tile_dgrad(const float* __restrict__ A, int lda,
                                           const float* __restrict__ B, int ldb,
                                           int K, int mt, int nt,
                                           const float* __restrict__ Zc,
                                           float* __restrict__ DZ, int ldc) {
  int lane = threadIdx.x & 31;
  int mr = lane & 15;
  int kh = (lane >> 4) << 1;
  const float* pA = A + (size_t)(mt * 16 + mr) * lda;
  int col = nt * 16 + mr;
  v8f acc = {};
  for (int k = 0; k < K; k += 4) {
    v2f a, b;
    a.x = pA[k + kh]; a.y = pA[k + kh + 1];
    b.x = B[(size_t)(k + kh) * ldb + col];
    b.y = B[(size_t)(k + kh + 1) * ldb + col];
    acc = wmma4(a, b, acc);
  }
  int rb = mt * 16 + ((lane >> 4) << 3);
#pragma unroll
  for (int i = 0; i < 8; ++i) {
    int row = rb + i;
    DZ[(size_t)row * ldc + col] = acc[i] * dsilu(Zc[(size_t)row * ldc + col]);
  }
}

// wgrad tile from global: C[i,j] = sum_{m<64} A[m,i]*B[m,j].
// Returns this lane's partial sum-of-squares of the written elements.
__device__ __forceinline__ float tile_wgrad_g(const float* __restrict__ A,
                                              int lda,
                                              const float* __restrict__ B,
                                              int ldb, int it, int jt,
                                              float* __restrict__ C, int ldc) {
  int lane = threadIdx.x & 31;
  int mr = lane & 15;
  int kh = (lane >> 4) << 1;
  int ai = it * 16 + mr;
  int bj = jt * 16 + mr;
  v8f acc = {};
  for (int k = 0; k < CHK; k += 4) {
    v2f a, b;
    a.x = A[(size_t)(k + kh) * lda + ai];
    a.y = A[(size_t)(k + kh + 1) * lda + ai];
    b.x = B[(size_t)(k + kh) * ldb + bj];
    b.y = B[(size_t)(k + kh + 1) * ldb + bj];
    acc = wmma4(a, b, acc);
  }
  int col = jt * 16 + mr;
  int rb = it * 16 + ((lane >> 4) << 3);
  float ss = 0.f;
#pragma unroll
  for (int i = 0; i < 8; ++i) {
    float v = acc[i];
    C[(size_t)(rb + i) * ldc + col] = v;
    ss = fmaf(v, v, ss);
  }
  return ss;
}

// wgrad tile from LDS panels (ds loads feeding WMMA).
__device__ __forceinline__ float tile_wgrad_lds(const float* Ap, const float* Bp,
                                                int ti, int tj, int it, int jt,
                                                float* __restrict__ C,
                                                int ldc) {
  int lane = threadIdx.x & 31;
  int mr = lane & 15;
  int kh = (lane >> 4) << 1;
  v8f acc = {};
#pragma unroll 4
  for (int k = 0; k < CHK; k += 4) {
    v2f a, b;
    a.x = Ap[(k + kh) * AP_LD + ti * 16 + mr];
    a.y = Ap[(k + kh + 1) * AP_LD + ti * 16 + mr];
    b.x = Bp[(k + kh) * BP_LD + tj * 16 + mr];
    b.y = Bp[(k + kh + 1) * BP_LD + tj * 16 + mr];
    acc = wmma4(a, b, acc);
  }
  int col = jt * 16 + mr;
  int rb = it * 16 + ((lane >> 4) << 3);
  float ss = 0.f;
#pragma unroll
  for (int i = 0; i < 8; ++i) {
    float v = acc[i];
    C[(size_t)(rb + i) * ldc + col] = v;
    ss = fmaf(v, v, ss);
  }
  return ss;
}

__global__ void nm_init(unsigned* bar) {
  bar[0] = 0u;
  bar[1] = 0u;
}

__global__ __launch_bounds__(NTHR) void nm_persistent(
    const float* __restrict__ x, const float* __restrict__ keys,
    const float* __restrict__ values, const float* __restrict__ tw,
    const float* __restrict__ tb, const float* __restrict__ ew,
    const float* __restrict__ eb, const float* __restrict__ aw,
    const float* __restrict__ ab, float* __restrict__ out,
    float* __restrict__ ws, unsigned* bar) {
  const int tid = threadIdx.x;
  const int gtid = blockIdx.x * NTHR + tid;
  const int wid = gtid >> 5;
  const int lane = tid & 31;
  const int wloc = tid >> 5;  // wave within block (0..7)
  unsigned lbar = 0;

  __shared__ float sred[NTHR];
  __shared__ float lds_a[CHK * AP_LD];  // 36864 B
  __shared__ float lds_b[CHK * BP_LD];  // 20480 B

  float* FLAT = ws + WS_FLAT;
  float* MOM  = ws + WS_MOM;
  float* GRAD = ws + WS_GRAD;
  float* Z0 = ws + WS_Z0;  float* H0 = ws + WS_H0;
  float* Z1 = ws + WS_Z1;  float* H1 = ws + WS_H1;
  float* Z2 = ws + WS_Z2;  float* H2 = ws + WS_H2;
  float* Y  = ws + WS_Y;   float* DY = ws + WS_DY;
  float* GU = ws + WS_GU;
  float* DZ2 = ws + WS_DZ2; float* DZ1 = ws + WS_DZ1; float* DZ0 = ws + WS_DZ0;
  float* PART = ws + WS_PART;

  for (int b = 0; b < NB; ++b) {
    // ---- batch init: carry MLP params, reset gamma=1 and momentum=0 ----
    if (b == 0)
      for (int i = gtid; i < P_MLP; i += NTOT) FLAT[i] = 0.f;
    if (gtid < HD) FLAT[P_MLP + gtid] = 1.f;
    for (int i = gtid; i < P_TOTAL; i += NTOT) MOM[i] = 0.f;
    grid_barrier(bar, &lbar);

    for (int c = 0; c < NCH; ++c) {
      const float* Kc = keys + (size_t)(b * 1024 + c * CHK) * HD;
      const float* Vc = values + (size_t)(b * 1024 + c * CHK) * HD;
      const float* xt = x + (size_t)(b * 1024 + c) * DM;

      // ---- 1: layer0 fwd ----
      for (int t = wid; t < 4 * 64; t += NWAVE)
        tile_fwd(Kc, HD, FLAT + OFF_W0, HD, HD, t >> 6, t & 63,
                 FLAT + OFF_B0, Z0, H0, HID);
      grid_barrier(bar, &lbar);

      // ---- 2/3: hidden layers ----
      for (int t = wid; t < 4 * 64; t += NWAVE)
        tile_fwd(H0, HID, FLAT + OFF_W1, HID, HID, t >> 6, t & 63,
                 FLAT + OFF_B1, Z1, H1, HID);
      grid_barrier(bar, &lbar);
      for (int t = wid; t < 4 * 64; t += NWAVE)
        tile_fwd(H1, HID, FLAT + OFF_W2, HID, HID, t >> 6, t & 63,
                 FLAT + OFF_B2, Z2, H2, HID);
      grid_barrier(bar, &lbar);

      // ---- 4: output + residual ----
      for (int t = wid; t < 4 * 16; t += NWAVE)
        tile_fwd_out(H2, HID, FLAT + OFF_WO, HID, HID, t >> 4, t & 15,
                     FLAT + OFF_BO, Kc, HD, Y, HD);
      grid_barrier(bar, &lbar);

      // ---- 5: RMSNorm fwd+bwd per row (block m == row m) ----
      {
        int m = blockIdx.x;
        float yv = Y[(size_t)m * HD + tid];
        float ss = block_reduce(yv * yv, sred);
        float r = rsqrtf(ss * (1.f / HD) + EPSF);
        float u = yv * r;
        float g = FLAT[OFF_G + tid];
        float dpred = (2.f / HD) * (g * u - Vc[(size_t)m * HD + tid]);
        float du = dpred * g;
        float cm = block_reduce(du * u, sred) * (1.f / HD);
        DY[(size_t)m * HD + tid] = r * (du - u * cm);
        GU[(size_t)m * HD + tid] = dpred * u;
      }
      grid_barrier(bar, &lbar);

      // ---- 6: dz2=(DY@WO)*silu'(Z2); gWO=DY^T@H2; gBO; gGamma ----
      for (int t = wid; t < 256 + 1024; t += NWAVE) {
        if (t < 256)
          tile_dgrad(DY, HD, FLAT + OFF_WO, HID, HD, t >> 6, t & 63, Z2, DZ2,
                     HID);
        else {
          int tt = t - 256;
          float ss = wave_sum(tile_wgrad_g(DY, HD, H2, HID, tt >> 6, tt & 63,
                                           GRAD + OFF_WO, HID));
          if (lane == 0) PART[SL_WO + tt] = ss;
        }
      }
      if (gtid < HD) {
        float s = 0.f;
        for (int m = 0; m < CHK; ++m) s += DY[(size_t)m * HD + gtid];
        GRAD[OFF_BO + gtid] = s;
        PART[SL_BO + gtid] = s * s;
      } else if (gtid < 2 * HD) {
        int n = gtid - HD;
        float s = 0.f;
        for (int m = 0; m < CHK; ++m) s += GU[(size_t)m * HD + n];
        GRAD[OFF_G + n] = s;
        PART[SL_G + n] = s * s;
      }
      grid_barrier(bar, &lbar);

      // ---- 7: gW2 = DZ2^T @ H1 via LDS panels; dz1; gb2 ----
      for (int p = 0; p < 2; ++p) {
        int patch = blockIdx.x + NBLK * p;  // 0..127
        int pi = patch >> 4;                // 8 row-groups of 128
        int pj = patch & 15;                // 16 col-groups of 64
        __syncthreads();
        for (int f = tid; f < 2048; f += NTHR) {  // A panel 64x128
          int row = f >> 5, c4 = (f & 31) << 2;
          async_g2l_b128(&lds_a[row * AP_LD + c4],
                         DZ2 + (size_t)row * HID + pi * 128 + c4);
        }
        for (int f = tid; f < 1024; f += NTHR) {  // B panel 64x64
          int row = f >> 4, c4 = (f & 15) << 2;
          async_g2l_b128(&lds_b[row * BP_LD + c4],
                         H1 + (size_t)row * HID + pj * 64 + c4);
        }
        wait_async0();
        __syncthreads();
        for (int t = wloc; t < 32; t += 8) {
          int ti = t >> 2, tj = t & 3;
          float ss = tile_wgrad_lds(lds_a, lds_b, ti, tj, pi * 8 + ti,
                                    pj * 4 + tj, GRAD + OFF_W2, HID);
          ss = wave_sum(ss);
          if (lane == 0) PART[SL_W2 + patch * 32 + t] = ss;
        }
      }
      if (wloc < 4) {
        int t = blockIdx.x * 4 + wloc;  // 0..255
        tile_dgrad(DZ2, HID, FLAT + OFF_W2, HID, HID, t >> 6, t & 63, Z1, DZ1,
                   HID);
      }
      if (gtid < HID) {
        float s = 0.f;
        for (int m = 0; m < CHK; ++m) s += DZ2[(size_t)m * HID + gtid];
        GRAD[OFF_B2 + gtid] = s;
        PART[SL_B2 + gtid] = s * s;
      }
      grid_barrier(bar, &lbar);

      // ---- 8: gW1 = DZ1^T @ H0 via LDS panels; dz0; gb1 ----
      for (int p = 0; p < 2; ++p) {
        int patch = blockIdx.x + NBLK * p;
        int pi = patch >> 4;
        int pj = patch & 15;
        __syncthreads();
        for (int f = tid; f < 2048; f += NTHR) {
          int row = f >> 5, c4 = (f & 31) << 2;
          async_g2l_b128(&lds_a[row * AP_LD + c4],
                         DZ1 + (size_t)row * HID + pi * 128 + c4);
        }
        for (int f = tid; f < 1024; f += NTHR) {
          int row = f >> 4, c4 = (f & 15) << 2;
          async_g2l_b128(&lds_b[row * BP_LD + c4],
                         H0 + (size_t)row * HID + pj * 64 + c4);
        }
        wait_async0();
        __syncthreads();
        for (int t = wloc; t < 32; t += 8) {
          int ti = t >> 2, tj = t & 3;
          float ss = tile_wgrad_lds(lds_a, lds_b, ti, tj, pi * 8 + ti,
                                    pj * 4 + tj, GRAD + OFF_W1, HID);
          ss = wave_sum(ss);
          if (lane == 0) PART[SL_W1 + patch * 32 + t] = ss;
        }
      }
      if (wloc < 4) {
        int t = blockIdx.x * 4 + wloc;
        tile_dgrad(DZ1, HID, FLAT + OFF_W1, HID, HID, t >> 6, t & 63, Z0, DZ0,
                   HID);
      }
      if (gtid < HID) {
        float s = 0.f;
        for (int m = 0; m < CHK; ++m) s += DZ1[(size_t)m * HID + gtid];
        GRAD[OFF_B1 + gtid] = s;
        PART[SL_B1 + gtid] = s * s;
      }
      grid_barrier(bar, &lbar);

      // ---- 9: gW0 = DZ0^T @ Kc; gb0 ----
      for (int t = wid; t < 1024; t += NWAVE) {
        float ss = wave_sum(tile_wgrad_g(DZ0, HID, Kc, HD, t >> 4, t & 15,
                                         GRAD + OFF_W0, HD));
        if (lane == 0) PART[SL_W0 + t] = ss;
      }
      if (gtid < HID) {
        float s = 0.f;
        for (int m = 0; m < CHK; ++m) s += DZ0[(size_t)m * HID + gtid];
        GRAD[OFF_B0 + gtid] = s;
        PART[SL_B0 + gtid] = s * s;
      }
      grid_barrier(bar, &lbar);

      // ---- 10: redundant per-block norm + gates, then update ----
      {
        float s = 0.f;
        for (int i = tid; i < NPART; i += NTHR) s += PART[i];
        float tot = block_reduce(s, sred);
        float scale = fminf(1.0f / (sqrtf(tot) + EPSF), 1.0f);
        float p0 = 0.f, p1 = 0.f, p2 = 0.f;
        for (int i = tid; i < DM; i += NTHR) {
          float xv = xt[i];
          p0 = fmaf(tw[i], xv, p0);
          p1 = fmaf(ew[i], xv, p1);
          p2 = fmaf(aw[i], xv, p2);
        }
        float th = fminf(sigm(block_reduce(p0, sred) + tb[0]), 0.1f);
        float et = sigm(block_reduce(p1, sred) + eb[0]);
        float al = sigm(block_reduce(p2, sred) + ab[0]);
        for (int i = gtid; i < P_TOTAL; i += NTOT) {
          float g = GRAD[i] * scale;
          float mm = fmaf(et, MOM[i], -th * g);
          MOM[i] = mm;
          FLAT[i] = fmaf(1.f - al, FLAT[i], mm);
        }
      }
      grid_barrier(bar, &lbar);
    }  // chunk loop

    // ---- emit (flats[b], moms[b]) before next batch's gamma/mom reset ----
    for (int i = gtid; i < P_TOTAL; i += NTOT) {
      out[(size_t)b * P_TOTAL + i] = FLAT[i];
      out[(size_t)(NB + b) * P_TOTAL + i] = MOM[i];
    }
    grid_barrier(bar, &lbar);
  }  // batch loop
}

extern "C" void kernel_launch(void* const* d_in, const int* in_sizes, int n_in,
                              void* d_out, int out_size, void* d_ws,
                              size_t ws_size, hipStream_t stream) {
  (void)in_sizes; (void)n_in; (void)out_size; (void)ws_size;
  const float* x    = (const float*)d_in[0];
  const float* keys = (const float*)d_in[1];
  const float* vals = (const float*)d_in[2];
  const float* tw   = (const float*)d_in[3];
  const float* tb   = (const float*)d_in[4];
  const float* ew   = (const float*)d_in[5];
  const float* eb   = (const float*)d_in[6];
  const float* aw   = (const float*)d_in[7];
  const float* ab   = (const float*)d_in[8];
  float* out = (float*)d_out;
  unsigned* bar = (unsigned*)d_ws;                  // 256B header
  float* wsF = (float*)((char*)d_ws + 256);

  nm_init<<<1, 1, 0, stream>>>(bar);
  nm_persistent<<<NBLK, NTHR, 0, stream>>>(x, keys, vals, tw, tb, ew, eb, aw,
                                           ab, out, wsF, bar);
}